// GeometryModule_51402168599381
// MI455X (gfx1250) — compile-verified
//
#include <hip/hip_runtime.h>
#include <hip/hip_bf16.h>

// ---------------------------------------------------------------------------
// Geometry normals + tangents for MI455X (gfx1250).
// Memory-bound scatter/gather workload: ~0.3 GFLOP vs ~200MB moved -> the
// win is data movement, not matrix ops (WMMA inapplicable: no contraction).
// CDNA5 features used:
//   * global_load_async_to_lds_b32 (ASYNCcnt) double-buffered, wave-local
//     staging of the streamed face/uv index tiles, with TH=NT so the 96MB
//     one-shot index stream does not evict the L2-hot gather/atomic set.
//   * s_wait_asynccnt pipelining (async loads complete in order).
//   * f32 global atomics into an L2-resident (88MB << 192MB) region.
//   * non-temporal final stores (output written once, never re-read).
// ---------------------------------------------------------------------------

#define NRM_EPS 1e-20f
#define DEN_EPS 1e-06f
#define TILE 256
#define MAX_BLOCKS 4096
#define CPOL_TH_NT 1  // gfx12-family CPol: TH bits [2:0], TH_NT = 1

typedef __attribute__((address_space(1))) int g1_int;  // global int
typedef __attribute__((address_space(3))) int l3_int;  // LDS int

__device__ __forceinline__ void cbar() { asm volatile("" ::: "memory"); }

__device__ __forceinline__ void async_ld_dword(const int* g, int* lds) {
#if __has_builtin(__builtin_amdgcn_global_load_async_to_lds_b32)
  __builtin_amdgcn_global_load_async_to_lds_b32((g1_int*)g, (l3_int*)lds, 0,
                                                CPOL_TH_NT);
#else
  *lds = *g;  // fallback: plain global load + ds store
#endif
}

__device__ __forceinline__ void wait_async_le0() {
#if __has_builtin(__builtin_amdgcn_global_load_async_to_lds_b32)
# if __has_builtin(__builtin_amdgcn_s_wait_asynccnt)
  __builtin_amdgcn_s_wait_asynccnt(0);
# else
  asm volatile("s_wait_asynccnt 0x0" ::: "memory");
# endif
#endif
}

__device__ __forceinline__ void wait_async_le6() {
#if __has_builtin(__builtin_amdgcn_global_load_async_to_lds_b32)
# if __has_builtin(__builtin_amdgcn_s_wait_asynccnt)
  __builtin_amdgcn_s_wait_asynccnt(6);
# else
  asm volatile("s_wait_asynccnt 0x6" ::: "memory");
# endif
#endif
}

// ---------------------------------------------------------------------------
// Kernel 1: zero the accumulator/output region (re-read by atomics -> RT).
// ---------------------------------------------------------------------------
__global__ void geo_zero(float* __restrict__ out, int n) {
  int i = blockIdx.x * blockDim.x + threadIdx.x;
  if (i < n) out[i] = 0.0f;
}

// ---------------------------------------------------------------------------
// Kernel 2: per-face accumulate (face normals + tangents) with async-staged
// index tiles. Each thread handles one face per tile; each thread's 6 index
// dwords (3 face ids + 3 uv ids) are staged by its own wave -> no barriers.
// ---------------------------------------------------------------------------
__global__ void __launch_bounds__(TILE) geo_face_accum(
    const float* __restrict__ pos,     // V x 3
    const float* __restrict__ tex,     // T x 2
    const int*   __restrict__ faces,   // F x 3
    const int*   __restrict__ uvfaces, // F x 3
    float* __restrict__ accN,          // V x 3 (pre-zeroed)
    float* __restrict__ accT,          // V x 3 (pre-zeroed)
    int F, int numTiles) {
  __shared__ int sF[2][TILE * 3];
  __shared__ int sU[2][TILE * 3];
  const int tid = threadIdx.x;

  auto stage = [&](int tile, int buf) {
    int f = tile * TILE + tid;
    if (f < F) {
      int base = tile * (TILE * 3) + 3 * tid;  // dword index into faces/uv
      int o = 3 * tid;
      async_ld_dword(faces   + base + 0, &sF[buf][o + 0]);
      async_ld_dword(faces   + base + 1, &sF[buf][o + 1]);
      async_ld_dword(faces   + base + 2, &sF[buf][o + 2]);
      async_ld_dword(uvfaces + base + 0, &sU[buf][o + 0]);
      async_ld_dword(uvfaces + base + 1, &sU[buf][o + 1]);
      async_ld_dword(uvfaces + base + 2, &sU[buf][o + 2]);
    }
  };

  int tile = blockIdx.x;
  if (tile >= numTiles) return;
  int buf = 0;
  stage(tile, buf);

  while (true) {
    int  next    = tile + (int)gridDim.x;
    bool hasNext = next < numTiles;
    if (hasNext) stage(next, buf ^ 1);
    // Async loads complete in order: waiting to <=6 guarantees the current
    // tile's 6 loads (this wave) are done even with the next tile in flight.
    if (hasNext) wait_async_le6(); else wait_async_le0();
    cbar();

    int f = tile * TILE + tid;
    if (f < F) {
      int o  = 3 * tid;
      int i0 = sF[buf][o + 0], i1 = sF[buf][o + 1], i2 = sF[buf][o + 2];
      int j0 = sU[buf][o + 0], j1 = sU[buf][o + 1], j2 = sU[buf][o + 2];

      const float* p0 = pos + 3 * (long long)i0;
      const float* p1 = pos + 3 * (long long)i1;
      const float* p2 = pos + 3 * (long long)i2;
      float ax = p0[0], ay = p0[1], az = p0[2];
      float e1x = p1[0] - ax, e1y = p1[1] - ay, e1z = p1[2] - az;
      float e2x = p2[0] - ax, e2y = p2[1] - ay, e2z = p2[2] - az;

      // face normal = cross(e1, e2)
      float nx = e1y * e2z - e1z * e2y;
      float ny = e1z * e2x - e1x * e2z;
      float nz = e1x * e2y - e1y * e2x;

      // tangent: uv edges
      const float* t0 = tex + 2 * (long long)j0;
      const float* t1 = tex + 2 * (long long)j1;
      const float* t2 = tex + 2 * (long long)j2;
      float u1x = t1[0] - t0[0], u1y = t1[1] - t0[1];
      float u2x = t2[0] - t0[0], u2y = t2[1] - t0[1];

      float denom = u1x * u2y - u1y * u2x;
      float sd = (denom > 0.0f) ? fmaxf(denom, DEN_EPS)
                                : fminf(denom, -DEN_EPS);
      float inv = 1.0f / sd;
      float tx = (e1x * u2y - e2x * u1y) * inv;
      float ty = (e1y * u2y - e2y * u1y) * inv;
      float tz = (e1z * u2y - e2z * u1y) * inv;

      // scatter-add (L2-resident accumulators)
      atomicAdd(&accN[3 * (long long)i0 + 0], nx);
      atomicAdd(&accN[3 * (long long)i0 + 1], ny);
      atomicAdd(&accN[3 * (long long)i0 + 2], nz);
      atomicAdd(&accN[3 * (long long)i1 + 0], nx);
      atomicAdd(&accN[3 * (long long)i1 + 1], ny);
      atomicAdd(&accN[3 * (long long)i1 + 2], nz);
      atomicAdd(&accN[3 * (long long)i2 + 0], nx);
      atomicAdd(&accN[3 * (long long)i2 + 1], ny);
      atomicAdd(&accN[3 * (long long)i2 + 2], nz);

      atomicAdd(&accT[3 * (long long)i0 + 0], tx);
      atomicAdd(&accT[3 * (long long)i0 + 1], ty);
      atomicAdd(&accT[3 * (long long)i0 + 2], tz);
      atomicAdd(&accT[3 * (long long)i1 + 0], tx);
      atomicAdd(&accT[3 * (long long)i1 + 1], ty);
      atomicAdd(&accT[3 * (long long)i1 + 2], tz);
      atomicAdd(&accT[3 * (long long)i2 + 0], tx);
      atomicAdd(&accT[3 * (long long)i2 + 1], ty);
      atomicAdd(&accT[3 * (long long)i2 + 2], tz);
    }

    if (!hasNext) break;
    tile = next;
    buf ^= 1;
  }
}

// ---------------------------------------------------------------------------
// Kernel 3: per-vertex finalize. Matches reference exactly:
//   n = where(dot(n,n) > eps, n, z);  n = n / sqrt(max(dot,eps))
//   t = t / sqrt(max(dot,eps)); t -= dot(t,n)*n; t = t / sqrt(max(dot,eps))
// Output is written once and never re-read -> non-temporal stores.
// ---------------------------------------------------------------------------
__global__ void geo_finalize(float* __restrict__ out, int V) {
  int v = blockIdx.x * blockDim.x + threadIdx.x;
  if (v >= V) return;
  float* n = out + 3 * (long long)v;
  float* t = out + 3 * (long long)V + 3 * (long long)v;

  float nx = n[0], ny = n[1], nz = n[2];
  float d = nx * nx + ny * ny + nz * nz;
  if (!(d > NRM_EPS)) { nx = 0.0f; ny = 0.0f; nz = 1.0f; d = 1.0f; }
  float r = rsqrtf(fmaxf(d, NRM_EPS));
  nx *= r; ny *= r; nz *= r;

  float tx = t[0], ty = t[1], tz = t[2];
  float dt = tx * tx + ty * ty + tz * tz;
  float rt = rsqrtf(fmaxf(dt, NRM_EPS));
  tx *= rt; ty *= rt; tz *= rt;

  float dp = tx * nx + ty * ny + tz * nz;
  tx -= dp * nx; ty -= dp * ny; tz -= dp * nz;
  dt = tx * tx + ty * ty + tz * tz;
  rt = rsqrtf(fmaxf(dt, NRM_EPS));
  tx *= rt; ty *= rt; tz *= rt;

  __builtin_nontemporal_store(nx, n + 0);
  __builtin_nontemporal_store(ny, n + 1);
  __builtin_nontemporal_store(nz, n + 2);
  __builtin_nontemporal_store(tx, t + 0);
  __builtin_nontemporal_store(ty, t + 1);
  __builtin_nontemporal_store(tz, t + 2);
}

// ---------------------------------------------------------------------------
extern "C" void kernel_launch(void* const* d_in, const int* in_sizes, int n_in,
                              void* d_out, int out_size, void* d_ws, size_t ws_size,
                              hipStream_t stream) {
  const float* pos     = (const float*)d_in[0];  // V*3 f32
  const float* tex     = (const float*)d_in[1];  // T*2 f32
  const int*   faces   = (const int*)d_in[2];    // F*3 int
  const int*   uvfaces = (const int*)d_in[3];    // F*3 int
  float*       out     = (float*)d_out;          // [nrm V*3 | tng V*3]

  const int V = in_sizes[0] / 3;
  const int F = in_sizes[2] / 3;

  // 1) zero accumulators (in-place in d_out) — re-done every call for
  //    determinism against the harness's poison-once policy.
  const int total = V * 6;
  geo_zero<<<(total + 255) / 256, 256, 0, stream>>>(out, total);

  // 2) face-parallel accumulate with async double-buffered index staging.
  const int numTiles = (F + TILE - 1) / TILE;
  int blocks = numTiles < MAX_BLOCKS ? numTiles : MAX_BLOCKS;
  geo_face_accum<<<blocks, TILE, 0, stream>>>(
      pos, tex, faces, uvfaces, out, out + (size_t)3 * V, F, numTiles);

  // 3) vertex-parallel finalize.
  geo_finalize<<<(V + 255) / 256, 256, 0, stream>>>(out, V);
}